// GCN_34574486733151
// MI455X (gfx1250) — compile-verified
//
#include <hip/hip_runtime.h>
#include <math.h>

typedef float v2f __attribute__((ext_vector_type(2)));
typedef float v8f __attribute__((ext_vector_type(8)));

#define HID 128
#define RTILES 4   // row tiles (16 rows each) per block; B fragments register-cached

static __device__ __forceinline__ void atomAddF(float* p, float v) {
    unsafeAtomicAdd(p, v);   // lowers to global_atomic_add_f32 on gfx1250
}

// ---------- degree / normalization ----------
__global__ __launch_bounds__(256) void k_init_dis(float* dis, int n) {
    int i = blockIdx.x * 256 + threadIdx.x;
    if (i < n) dis[i] = 1.0f;   // self-loop contributes 1 to degree
}

__global__ __launch_bounds__(256) void k_count_deg(const int* __restrict__ col,
                                                   float* __restrict__ dis, int e) {
    int i = blockIdx.x * 256 + threadIdx.x;
    if (i < e) atomAddF(&dis[col[i]], 1.0f);
}

__global__ __launch_bounds__(256) void k_finish_dis(float* dis, int n) {
    int i = blockIdx.x * 256 + threadIdx.x;
    if (i < n) dis[i] = rsqrtf(dis[i]);   // deg >= 1 always (self loops)
}

// ---------- f32 WMMA GEMM: C[nrows x 128] = act(A)[nrows x 128] * W[128 x 128] ----------
// 8 waves/block = 8 column tiles; each wave register-caches its 128x16 W strip as
// 32 v2f fragments and sweeps RTILES row tiles. Optional fused epilogue writes
// Hinit = dis[row]^2 * C + bias (GCN self-loop + bias term).
template <bool RELU_A, bool FUSE_INIT>
__global__ __launch_bounds__(256) void k_gemm_wmma(const float* __restrict__ A,
                                                   const float* __restrict__ W,
                                                   float* __restrict__ C,
                                                   float* __restrict__ Hinit,
                                                   const float* __restrict__ dis,
                                                   const float* __restrict__ bias,
                                                   int nrows) {
    int rowBase0 = blockIdx.x * (16 * RTILES);
    if (rowBase0 >= nrows) return;                // block-uniform
    int wave = threadIdx.x >> 5;                  // 0..7 -> column tile
    int lane = threadIdx.x & 31;
    int colBase = wave * 16;
    int m  = lane & 15;                           // M index (A rows) / N index (B,C cols)
    int kb = (lane >> 4) * 2;                     // half-wave selects K pair
    int mb = (lane >> 4) * 8;

    // register-cache the whole 128x16 weight strip for this column tile
    v2f bfrag[32];
    #pragma unroll
    for (int kk = 0; kk < 32; ++kk) {
        int k0 = kk * 4;
        bfrag[kk].x = W[(size_t)(k0 + kb)     * HID + colBase + m];
        bfrag[kk].y = W[(size_t)(k0 + kb + 1) * HID + colBase + m];
    }

    float bcol = 0.0f;
    if (FUSE_INIT) bcol = bias[colBase + m];

    #pragma unroll
    for (int r = 0; r < RTILES; ++r) {
        int rowBase = rowBase0 + r * 16;
        if (rowBase >= nrows) break;              // block-uniform (nrows % 16 == 0)
        const float* arow = A + (size_t)(rowBase + m) * HID;
        v8f acc = {};
        #pragma unroll
        for (int kk = 0; kk < 32; ++kk) {
            v2f a;
            if (RELU_A) {
                a.x = fmaxf(arow[kk * 4 + kb], 0.0f);
                a.y = fmaxf(arow[kk * 4 + kb + 1], 0.0f);
            } else {
                a.x = arow[kk * 4 + kb];
                a.y = arow[kk * 4 + kb + 1];
            }
            acc = __builtin_amdgcn_wmma_f32_16x16x4_f32(false, a, false, bfrag[kk],
                                                        (short)0, acc, false, false);
        }
        // C/D layout: VGPR v holds row (v + 8*(lane>>4)), col = lane&15
        #pragma unroll
        for (int v = 0; v < 8; ++v) {
            int rr = rowBase + mb + v;
            float val = acc[v];
            C[(size_t)rr * HID + colBase + m] = val;
            if (FUSE_INIT) {
                float s = dis[rr];
                Hinit[(size_t)rr * HID + colBase + m] = val * s * s + bcol;
            }
        }
    }
}

// ---------- h = dis^2 * xw + bias  (layer-2 init; separate pass to avoid in-place race) ----------
__global__ __launch_bounds__(256) void k_init_h(const float* __restrict__ xw,
                                                const float* __restrict__ dis,
                                                const float* __restrict__ bias,
                                                float* __restrict__ h, int n) {
    int i = blockIdx.x * 256 + threadIdx.x;
    int node = i >> 5, q = i & 31;
    if (node >= n) return;
    float s = dis[node]; s = s * s;
    float4 v = ((const float4*)(xw + (size_t)node * HID))[q];
    float4 b = ((const float4*)bias)[q];
    float4 o = { v.x * s + b.x, v.y * s + b.y, v.z * s + b.z, v.w * s + b.w };
    ((float4*)(h + (size_t)node * HID))[q] = o;
}

// ---------- edge scatter: h[col] += dis[row]*dis[col] * xw[row]  (one wave / edge) ----------
__global__ __launch_bounds__(256) void k_scatter(const float* __restrict__ xw,
                                                 const float* __restrict__ dis,
                                                 const int* __restrict__ row,
                                                 const int* __restrict__ col,
                                                 float* __restrict__ h, int e) {
    int gid  = blockIdx.x * 256 + threadIdx.x;
    int edge = gid >> 5, lane = gid & 31;
    if (edge >= e) return;
    int r = row[edge], c = col[edge];
    float nrm = dis[r] * dis[c];
    float4 v = ((const float4*)(xw + (size_t)r * HID))[lane];
    float* dst = h + (size_t)c * HID + lane * 4;
    atomAddF(dst + 0, nrm * v.x);
    atomAddF(dst + 1, nrm * v.y);
    atomAddF(dst + 2, nrm * v.z);
    atomAddF(dst + 3, nrm * v.w);
}

__global__ __launch_bounds__(256) void k_zero(float* p, int n) {
    int i = blockIdx.x * 256 + threadIdx.x;
    if (i < n) p[i] = 0.0f;
}

// ---------- mean pool with fused ReLU on loads ----------
__global__ __launch_bounds__(256) void k_pool(const float* __restrict__ h,
                                              const int* __restrict__ batch,
                                              float* __restrict__ pooled,
                                              float* __restrict__ counts, int n) {
    int i = blockIdx.x * 256 + threadIdx.x;
    int node = i >> 5, q = i & 31;
    if (node >= n) return;
    int g = batch[node];
    float4 v = ((const float4*)(h + (size_t)node * HID))[q];
    float* dst = pooled + (size_t)g * HID + q * 4;
    atomAddF(dst + 0, fmaxf(v.x, 0.0f));
    atomAddF(dst + 1, fmaxf(v.y, 0.0f));
    atomAddF(dst + 2, fmaxf(v.z, 0.0f));
    atomAddF(dst + 3, fmaxf(v.w, 0.0f));
    if (q == 0) atomAddF(&counts[g], 1.0f);
}

// ---------- head: logits = mean_pool @ wf + bf ; log_softmax over 16 ----------
__global__ __launch_bounds__(32) void k_head(const float* __restrict__ pooled,
                                             const float* __restrict__ counts,
                                             const float* __restrict__ wf,
                                             const float* __restrict__ bf,
                                             float* __restrict__ out) {
    int g = blockIdx.x, lane = threadIdx.x;
    __shared__ float p[HID];
    float inv = 1.0f / fmaxf(counts[g], 1.0f);
    for (int i = lane; i < HID; i += 32)
        p[i] = pooled[(size_t)g * HID + i] * inv;
    __syncthreads();
    float logit = -INFINITY;
    if (lane < 16) {
        float acc = bf[lane];
        for (int k = 0; k < HID; ++k) acc += p[k] * wf[k * 16 + lane];
        logit = acc;
    }
    float m = logit;
    #pragma unroll
    for (int off = 16; off > 0; off >>= 1) m = fmaxf(m, __shfl_xor(m, off, 32));
    float ex = (lane < 16) ? expf(logit - m) : 0.0f;
    float s = ex;
    #pragma unroll
    for (int off = 16; off > 0; off >>= 1) s += __shfl_xor(s, off, 32);
    if (lane < 16) out[(size_t)g * 16 + lane] = (logit - m) - logf(s);
}

extern "C" void kernel_launch(void* const* d_in, const int* in_sizes, int n_in,
                              void* d_out, int out_size, void* d_ws, size_t ws_size,
                              hipStream_t stream) {
    const float* x   = (const float*)d_in[0];
    const int*   ei  = (const int*)  d_in[1];
    const int*   bat = (const int*)  d_in[2];
    const float* w1  = (const float*)d_in[3];
    const float* b1  = (const float*)d_in[4];
    const float* w2  = (const float*)d_in[5];
    const float* b2  = (const float*)d_in[6];
    const float* wf  = (const float*)d_in[7];
    const float* bf  = (const float*)d_in[8];
    float* out = (float*)d_out;

    const int N = in_sizes[0] / HID;      // 100000
    const int E = in_sizes[1] / 2;        // 1600000
    const int G = out_size / 16;          // 1024
    const int* row = ei;
    const int* col = ei + E;

    // workspace layout (floats)
    float* ws = (float*)d_ws;
    size_t off = 0;
    float* dis    = ws + off; off += ((size_t)N + 511) & ~(size_t)511;
    float* xw     = ws + off; off += (size_t)N * HID;
    float* h      = ws + off; off += (size_t)N * HID;
    float* pooled = ws + off; off += (size_t)G * HID;
    float* counts = ws + off; off += (size_t)G;
    (void)ws_size; (void)n_in;

    const int nodeB  = (N + 255) / 256;
    const int edgeB  = (E + 255) / 256;
    const int quadB  = ((N * 32) + 255) / 256;         // one thread per float4 of features
    const int edgeWB = ((E * 32) + 255) / 256;         // one wave per edge
    const int gemmB  = (N + 16 * RTILES - 1) / (16 * RTILES);

    // normalization: dis = rsqrt(deg + 1)
    k_init_dis  <<<nodeB, 256, 0, stream>>>(dis, N);
    k_count_deg <<<edgeB, 256, 0, stream>>>(col, dis, E);
    k_finish_dis<<<nodeB, 256, 0, stream>>>(dis, N);

    // ---- layer 1: xw = x @ w1, fused h = dis^2*xw + b1 ----
    k_gemm_wmma<false, true><<<gemmB, 256, 0, stream>>>(x, w1, xw, h, dis, b1, N);
    k_scatter<<<edgeWB, 256, 0, stream>>>(xw, dis, row, col, h, E);

    // ---- layer 2: xw2 = relu(h) @ w2 (ReLU fused into A loads) ----
    k_gemm_wmma<true, false><<<gemmB, 256, 0, stream>>>(h, w2, xw, nullptr, nullptr, nullptr, N);
    k_init_h<<<quadB, 256, 0, stream>>>(xw, dis, b2, h, N);   // h consumed by gemm above
    k_scatter<<<edgeWB, 256, 0, stream>>>(xw, dis, row, col, h, E);

    // ---- mean pool (ReLU fused) + head ----
    k_zero<<<((G * HID + G) + 255) / 256, 256, 0, stream>>>(pooled, G * HID + G);
    k_pool<<<quadB, 256, 0, stream>>>(h, bat, pooled, counts, N);
    k_head<<<G, 32, 0, stream>>>(pooled, counts, wf, bf, out);
}